// StillingerWeberLayer_8349416423610
// MI455X (gfx1250) — compile-verified
//
#include <hip/hip_runtime.h>

// Stillinger-Weber energy for MI455X (gfx1250, wave32).
// One wave per atom. The per-atom 16x16 Gram matrix (einsum 'njc,nkc->njk')
// is computed with a single V_WMMA_F32_16X16X4_F32 (K=3 padded to 4).
// Fast TRANS-unit math (v_sqrt_f32 / v_rcp_f32) instead of IEEE div/sqrt
// expansions. Deterministic two-stage reduction (no float atomics).

#define KNBR 16
#define WAVES_PER_BLOCK 8
#define BLOCK_THREADS 256
#define NUM_BLOCKS 1024
#define EPSF 1e-12f

typedef float v2f __attribute__((ext_vector_type(2)));
typedef float v8f __attribute__((ext_vector_type(8)));

__device__ __forceinline__ float sel3(float v0, float v1, float v2, int i) {
  return (i == 0) ? v0 : ((i == 1) ? v1 : v2);
}
__device__ __forceinline__ float fast_sqrt(float x) {
  return __builtin_amdgcn_sqrtf(x);          // single v_sqrt_f32
}
__device__ __forceinline__ float fast_rcp(float x) {
  return __builtin_amdgcn_rcpf(x);           // single v_rcp_f32
}

__global__ __launch_bounds__(BLOCK_THREADS) void sw_energy_kernel(
    const float* __restrict__ coords,
    const float* __restrict__ Ap,   const float* __restrict__ Bp,
    const float* __restrict__ pp,   const float* __restrict__ qp,
    const float* __restrict__ sigp, const float* __restrict__ gamp,
    const float* __restrict__ cutp, const float* __restrict__ lamp,
    const float* __restrict__ cbp,  const float* __restrict__ cjkp,
    const int*  __restrict__ elements,
    const long long* __restrict__ nbr,
    float* __restrict__ partials, int N)
{
  __shared__ float s_rij[WAVES_PER_BLOCK][KNBR];
  __shared__ float s_sq [WAVES_PER_BLOCK][KNBR];
  __shared__ float s_cut[WAVES_PER_BLOCK][KNBR];
  __shared__ float s_gam[WAVES_PER_BLOCK][KNBR];
  __shared__ int   s_ej [WAVES_PER_BLOCK][KNBR];
  __shared__ float s_wsum[WAVES_PER_BLOCK];

  const int  lane = threadIdx.x & 31;
  const int  wid  = threadIdx.x >> 5;
  const int  m    = lane & 15;          // neighbor handled by this lane (k-column)
  const bool hi   = lane >= 16;
  const int  waveGlobal = blockIdx.x * WAVES_PER_BLOCK + wid;
  const int  totalWaves = gridDim.x * WAVES_PER_BLOCK;

  // Hoist tiny parameter tables into registers (L2-resident, wave-uniform).
  const float A0 = Ap[0],  A1 = Ap[1],  A2 = Ap[2];
  const float B0 = Bp[0],  B1 = Bp[1],  B2 = Bp[2];
  const float p0 = pp[0],  p1 = pp[1],  p2 = pp[2];
  const float q0 = qp[0],  q1 = qp[1],  q2 = qp[2];
  const float g0 = gamp[0], g1 = gamp[1], g2 = gamp[2];
  const float sg0 = sigp[0], sg1 = sigp[1], sg2 = sigp[2];
  const float ct0 = cutp[0], ct1 = cutp[1], ct2 = cutp[2];
  const float lam0 = lamp[0], lam1 = lamp[1];
  const float cb0 = cbp[0],   cb1 = cbp[1];
  const float cjk0 = cjkp[0], cjk1 = cjkp[1];

  float acc = 0.0f;

  for (int i = waveGlobal; i < N; i += totalWaves) {
    // ---- gather: lanes L and L+16 both handle neighbor m = L%16 ----
    const float xix = coords[3 * i + 0];
    const float xiy = coords[3 * i + 1];
    const float xiz = coords[3 * i + 2];
    const int   ei  = elements[i];

    const long long jdx = nbr[(long long)i * KNBR + m];
    const float xjx = coords[3 * jdx + 0];
    const float xjy = coords[3 * jdx + 1];
    const float xjz = coords[3 * jdx + 2];
    const int   ejv = elements[jdx];

    // ---- 2-body term (lanes 0-15 contribute; 16-31 mirror for k-side reuse) ----
    const float dx = xjx - xix, dy = xjy - xiy, dz = xjz - xiz;
    const float rij = fast_sqrt(fmaxf(dx * dx + dy * dy + dz * dz, EPSF));
    const int   ij  = ei + ejv;
    const float cut = sel3(ct0, ct1, ct2, ij);
    const bool  m2  = rij < cut;
    const float r2v = m2 ? rij : (cut - 1.0f);
    const float sig = sel3(sg0, sg1, sg2, ij);
    const float sr  = sig * fast_rcp(r2v);
    const float srp = __powf(sr, sel3(p0, p1, p2, ij));   // exp2/log2 fast path
    const float srq = __powf(sr, sel3(q0, q1, q2, ij));
    const float E2  = sel3(A0, A1, A2, ij) *
                      (sel3(B0, B1, B2, ij) * srp - srq) *
                      __expf(sig * fast_rcp(r2v - cut));
    acc += (!hi && m2) ? (0.5f * E2) : 0.0f;

    const float sqv = xjx * xjx + xjy * xjy + xjz * xjz;
    const float gam = sel3(g0, g1, g2, ij);

    // ---- stash per-neighbor scalars for j-side broadcasts ----
    s_rij[wid][m] = rij;
    s_sq [wid][m] = sqv;
    s_cut[wid][m] = cut;
    s_gam[wid][m] = gam;
    s_ej [wid][m] = ejv;
    __builtin_amdgcn_wave_barrier();   // order LDS stores before loads below

    // ---- Gram matrix via WMMA: D(j,k) = sum_c X(j,c)*X(k,c) ----
    // A layout (16x4 f32): VGPR0 = K0|K2 across lane halves, VGPR1 = K1|K3.
    // For the symmetric product X*X^T, B's register image equals A's.
    v2f a;
    a[0] = hi ? xjz : xjx;             // K=2 | K=0
    a[1] = hi ? 0.0f : xjy;            // K=3 (zero pad) | K=1
    v8f c = {};
    c = __builtin_amdgcn_wmma_f32_16x16x4_f32(
        /*neg_a=*/false, a, /*neg_b=*/false, a,
        /*c_mod=*/(short)0, c, /*reuse_a=*/false, /*reuse_b=*/false);

    // k-side scalars are this lane's own values (k = lane%16).
    const int   k    = m;
    const float rik  = rij;
    const float cutk = cut;
    const float gamk = gam;

    // ---- 3-body: 8 C-rows per lane; element (j = r + 8*hi, k = lane%16) ----
    float acc3 = 0.0f;
#pragma unroll
    for (int r = 0; r < 8; ++r) {
      const int   j    = r + (hi ? 8 : 0);
      const float rijj = s_rij[wid][j];
      const float sqj  = s_sq [wid][j];
      const float cutj = s_cut[wid][j];
      const float gamj = s_gam[wid][j];
      const int   ej   = s_ej [wid][j];

      const float dotjk = c[r];
      const float rjk   = fast_sqrt(fmaxf(sqj + sqv - 2.0f * dotjk, EPSF));

      const bool species_ok = (ei != ej) && (ej == ejv);
      int ijk = 2 - (ei + ej + ejv);
      ijk = ijk < 0 ? 0 : (ijk > 1 ? 1 : ijk);
      const float cjkc = ijk ? cjk1 : cjk0;

      const bool m3 = (j < k) && species_ok &&
                      (rijj < cutj) && (rik < cutk) && (rjk < cjkc);

      const float rs  = m3 ? rijj : (cutj - 1.0f);
      const float rks = m3 ? rik  : (cutk - 1.0f);
      const float cosv = (rs * rs + rks * rks - rjk * rjk) *
                         fast_rcp(2.0f * rs * rks);
      const float lamv = ijk ? lam1 : lam0;
      const float cbv  = ijk ? cb1  : cb0;
      const float dcb  = cosv - cbv;
      const float E3 = lamv *
                       __expf(gamj * fast_rcp(rs - cutj) +
                              gamk * fast_rcp(rks - cutk)) *
                       dcb * dcb;
      acc3 += m3 ? E3 : 0.0f;
    }
    acc += acc3;
  }

  // ---- wave reduce (wave32 shuffles) ----
  float v = acc;
#pragma unroll
  for (int off = 16; off > 0; off >>= 1)
    v += __shfl_xor(v, off, 32);
  if (lane == 0) s_wsum[wid] = v;
  __syncthreads();
  if (threadIdx.x == 0) {
    float t = 0.0f;
#pragma unroll
    for (int w = 0; w < WAVES_PER_BLOCK; ++w) t += s_wsum[w];
    partials[blockIdx.x] = t;
  }
}

__global__ __launch_bounds__(256) void sw_reduce_kernel(
    const float* __restrict__ partials, float* __restrict__ out, int nparts)
{
  __shared__ float s[256];
  float v = 0.0f;
  for (int i = threadIdx.x; i < nparts; i += 256) v += partials[i];
  s[threadIdx.x] = v;
  __syncthreads();
#pragma unroll
  for (int off = 128; off > 0; off >>= 1) {
    if ((int)threadIdx.x < off) s[threadIdx.x] += s[threadIdx.x + off];
    __syncthreads();
  }
  if (threadIdx.x == 0) out[0] = s[0];
}

extern "C" void kernel_launch(void* const* d_in, const int* in_sizes, int n_in,
                              void* d_out, int out_size, void* d_ws, size_t ws_size,
                              hipStream_t stream) {
  const float*     coords   = (const float*)d_in[0];
  const float*     Ap       = (const float*)d_in[1];
  const float*     Bp       = (const float*)d_in[2];
  const float*     pp       = (const float*)d_in[3];
  const float*     qp       = (const float*)d_in[4];
  const float*     sigp     = (const float*)d_in[5];
  const float*     gamp     = (const float*)d_in[6];
  const float*     cutp     = (const float*)d_in[7];
  const float*     lamp     = (const float*)d_in[8];
  const float*     cbp      = (const float*)d_in[9];
  const float*     cjkp     = (const float*)d_in[10];
  const int*       elements = (const int*)d_in[11];
  const long long* nbr      = (const long long*)d_in[12];
  const int N = in_sizes[11];

  float* partials = (float*)d_ws;   // NUM_BLOCKS floats of scratch

  sw_energy_kernel<<<NUM_BLOCKS, BLOCK_THREADS, 0, stream>>>(
      coords, Ap, Bp, pp, qp, sigp, gamp, cutp, lamp, cbp, cjkp,
      elements, nbr, partials, N);
  sw_reduce_kernel<<<1, 256, 0, stream>>>(partials, (float*)d_out, NUM_BLOCKS);
}